// FocalBCELoss_20469814133055
// MI455X (gfx1250) — compile-verified
//
#include <hip/hip_runtime.h>
#include <math.h>

typedef __attribute__((ext_vector_type(2))) float v2f;
typedef __attribute__((ext_vector_type(4))) float v4f;
typedef __attribute__((ext_vector_type(8))) float v8f;

#define NROWS 262144
#define NCOLS 64
#define GAMMA_IS_2 1

// ---------------------------------------------------------------------------
// Wave32 full-lane sum using the CDNA5 matrix pipe.
// A = ones (16x4 f32), B: VGPR0 = per-lane partial (rows K=0 from lanes 0-15,
// K=1 from lanes 16-31), VGPR1 = 0 (rows K=2,3).  D[m,n] = acc[n] + acc[n+16],
// identical for every row m, so after one v_wmma_f32_16x16x4_f32 each lane's
// d[0] holds one of 16 distinct pair-sums; 4 xor-shuffles finish the wave sum.
// Requires EXEC == all ones at the call site (guaranteed: no divergence).
// ---------------------------------------------------------------------------
__device__ __forceinline__ float wave_sum_wmma(float acc) {
    v2f a; a.x = 1.0f; a.y = 1.0f;   // A matrix: all ones
    v2f b; b.x = acc;  b.y = 0.0f;   // B matrix: lane partials in rows 0,1
    v8f c = {};
    c = __builtin_amdgcn_wmma_f32_16x16x4_f32(
            /*neg_a=*/false, a, /*neg_b=*/false, b,
            /*c_mod=*/(short)0, c, /*reuse_a=*/false, /*reuse_b=*/false);
    float s = c[0];
    s += __shfl_xor(s, 1, 32);
    s += __shfl_xor(s, 2, 32);
    s += __shfl_xor(s, 4, 32);
    s += __shfl_xor(s, 8, 32);
    return s;  // full wave sum, every lane
}

__device__ __forceinline__ float focal_term(float p, float t, float a) {
    // -( t*log(p)*(1-p)^2*alpha + (1-t)*log(1-p)*p^2 ),  gamma == 2
    float omp   = 1.0f - p;
    float left  = t * __logf(p) * (omp * omp) * a;
    float right = (1.0f - t) * __logf(omp) * (p * p);
    return -(left + right);
}

__global__ void focal_main(const float* __restrict__ inputs,
                           const float* __restrict__ targets,
                           const float* __restrict__ alpha,
                           float* __restrict__ partial,
                           int n4) {
    __shared__ float s_alpha[NCOLS];
    __shared__ float s_wave[8];

    if (threadIdx.x < NCOLS) s_alpha[threadIdx.x] = alpha[threadIdx.x];
    __syncthreads();

    const v4f* __restrict__ p4 = (const v4f*)inputs;
    const v4f* __restrict__ t4 = (const v4f*)targets;

    float acc = 0.0f;
    // 4 M float4s / (4096*256 threads) = 4 iterations, fully coalesced b128.
    for (int i = blockIdx.x * blockDim.x + threadIdx.x; i < n4;
         i += gridDim.x * blockDim.x) {
        // Streaming single-pass data: non-temporal loads (TH=NT) so we don't
        // pin 128 MiB of dead data in the cache hierarchy.
        v4f pv = __builtin_nontemporal_load(&p4[i]);
        v4f tv = __builtin_nontemporal_load(&t4[i]);
        int cbase = (i & (NCOLS / 4 - 1)) << 2;  // column of first element
        acc += focal_term(pv.x, tv.x, s_alpha[cbase + 0]);
        acc += focal_term(pv.y, tv.y, s_alpha[cbase + 1]);
        acc += focal_term(pv.z, tv.z, s_alpha[cbase + 2]);
        acc += focal_term(pv.w, tv.w, s_alpha[cbase + 3]);
    }

    // EXEC is all ones here (uniform loop exit): WMMA-legal.
    float wsum = wave_sum_wmma(acc);
    if ((threadIdx.x & 31) == 0) s_wave[threadIdx.x >> 5] = wsum;
    __syncthreads();

    if (threadIdx.x == 0) {
        float bsum = 0.0f;
        #pragma unroll
        for (int w = 0; w < 8; ++w) bsum += s_wave[w];
        partial[blockIdx.x] = bsum;
    }
}

__global__ void focal_final(const float* __restrict__ partial, int nparts,
                            float* __restrict__ out, float scale) {
    __shared__ float s_wave[8];

    float acc = 0.0f;
    for (int i = threadIdx.x; i < nparts; i += blockDim.x)
        acc += partial[i];

    float wsum = wave_sum_wmma(acc);
    if ((threadIdx.x & 31) == 0) s_wave[threadIdx.x >> 5] = wsum;
    __syncthreads();

    if (threadIdx.x == 0) {
        float total = 0.0f;
        #pragma unroll
        for (int w = 0; w < 8; ++w) total += s_wave[w];
        out[0] = total * scale;
    }
}

extern "C" void kernel_launch(void* const* d_in, const int* in_sizes, int n_in,
                              void* d_out, int out_size, void* d_ws, size_t ws_size,
                              hipStream_t stream) {
    const float* inputs  = (const float*)d_in[0];
    const float* targets = (const float*)d_in[1];
    const float* alpha   = (const float*)d_in[2];
    float* out     = (float*)d_out;
    float* partial = (float*)d_ws;  // 4096 floats = 16 KB, overwritten each call

    const int n4     = (NROWS * NCOLS) / 4;  // 4,194,304 float4 elements
    const int blocks = 4096;                 // 8 wave32 per block
    const float scale = 1.0f / ((float)NROWS * (float)NCOLS);

    focal_main<<<blocks, 256, 0, stream>>>(inputs, targets, alpha, partial, n4);
    focal_final<<<1, 256, 0, stream>>>(partial, blocks, out, scale);
}